// NetworkClass_13022340841819
// MI455X (gfx1250) — compile-verified
//
#include <hip/hip_runtime.h>

// LIF membrane update: v_new = alpha*v + sum_i x[b,i,n]*w[i,n] - vth*z ; z_new = (v_new - vth > 0)
// Memory-bound (0.5 FLOP/byte): implemented as per-n GEMV tiles on V_WMMA_F32_16X16X4_F32
// with x staged global->LDS (async path when available) for coalescing + transpose.

#define BDIM  128
#define NIN   1024
#define NNC   512
#define ALPHA 0.995f
#define VTH   2.0f

#define KI 32                    // i-chunk staged per iteration
#define BT 16                    // b tile (WMMA M)
#define NT 16                    // n tile per workgroup
#define BSTRIDE (KI * NT + 4)    // 516 dwords: bank-conflict pad (516%64==4) + keeps 16B align

typedef __attribute__((ext_vector_type(2))) float v2f;
typedef __attribute__((ext_vector_type(8))) float v8f;

#if __has_builtin(__builtin_amdgcn_global_load_async_to_lds_b128)
#define HAVE_ASYNC 1
typedef int v4i_vs __attribute__((vector_size(16)));            // matches builtin param elem type
typedef __attribute__((address_space(1))) v4i_vs gbl_v4i;       // AS1 (printed as __device__)
typedef __attribute__((address_space(3))) v4i_vs lds_v4i;       // AS3 (LDS)
__device__ __forceinline__ void async_b128(const void* g, void* l) {
    // AS3 ptr = low 32 bits of flat addr (ISA 10.2); AS1 = full 64-bit.
    __builtin_amdgcn_global_load_async_to_lds_b128(
        (gbl_v4i*)(unsigned long long)g,
        (lds_v4i*)(unsigned int)(unsigned long long)l, 0, 0);
}
__device__ __forceinline__ void wait_async0() {
#if __has_builtin(__builtin_amdgcn_s_wait_asynccnt)
    __builtin_amdgcn_s_wait_asynccnt(0);
#else
    asm volatile("s_wait_asynccnt 0" ::: "memory");
#endif
}
#else
#define HAVE_ASYNC 0
#endif

__global__ __launch_bounds__(128) void lif_wmma_kernel(
    const float* __restrict__ x, const float* __restrict__ w,
    const float* __restrict__ v, const float* __restrict__ z,
    float* __restrict__ outv, float* __restrict__ outz)
{
    __shared__ __align__(16) float xs[BT * BSTRIDE];  // 33 KB, padded [b][i][n]
    __shared__ __align__(16) float ws[KI * NT];       // 2 KB,  [i][n]

    const int tid  = threadIdx.x;
    const int lane = tid & 31;
    const int wave = tid >> 5;            // 0..3, each wave owns 4 n-values
    const int n0   = blockIdx.x * NT;
    const int b0   = blockIdx.y * BT;

    // A-fragment lane mapping (ISA 7.12.2, 32-bit A 16x4):
    //   lanes 0-15: M=lane, VGPR0/1 = K=0/1 ; lanes 16-31: M=lane-16, VGPR0/1 = K=2/3
    const int brow  = lane & 15;
    const int khalf = (lane >> 4) << 1;   // 0 or 2

    v8f acc[4] = {};                      // one 16x16 f32 accumulator per owned n

    for (int i0 = 0; i0 < NIN; i0 += KI) {
        __syncthreads();                  // previous chunk fully consumed

        // ---- stage x[b0:+16, i0:+KI, n0:+16] -> xs (2048 float4, 16 per thread)
        for (int r = 0; r < 16; ++r) {
            int linear = r * 128 + tid;   // 0..2047
            int n4  = linear & 3;
            int row = linear >> 2;        // 0..511 = b*KI + kk
            int kk  = row % KI;
            int b   = row / KI;
            const float* gp = x + ((size_t)(b0 + b) * NIN + (i0 + kk)) * NNC + n0 + n4 * 4;
            float*       lp = &xs[b * BSTRIDE + kk * NT + n4 * 4];
#if HAVE_ASYNC
            async_b128(gp, lp);
#else
            *(float4*)lp = *(const float4*)gp;
#endif
        }
        // ---- stage w[i0:+KI, n0:+16] -> ws (128 float4, 1 per thread)
        {
            int n4 = tid & 3;
            int kk = tid >> 2;            // 0..31
            const float* gp = w + (size_t)(i0 + kk) * NNC + n0 + n4 * 4;
            float*       lp = &ws[kk * NT + n4 * 4];
#if HAVE_ASYNC
            async_b128(gp, lp);
#else
            *(float4*)lp = *(const float4*)gp;
#endif
        }
#if HAVE_ASYNC
        wait_async0();
#endif
        __syncthreads();

        // ---- 8 k-steps x 4 n: D(16b x 16) += A(16b x 4i) * B(4i x 16, w broadcast across cols)
        for (int k = 0; k < KI / 4; ++k) {
            const int kk = (k << 2) + khalf;
#pragma unroll
            for (int nn = 0; nn < 4; ++nn) {
                const int nl = (wave << 2) + nn;
                v2f a, bf;
                a.x  = xs[brow * BSTRIDE + kk * NT + nl];        // conflict-free: stride 516
                a.y  = xs[brow * BSTRIDE + (kk + 1) * NT + nl];
                bf.x = ws[kk * NT + nl];                          // LDS broadcast per half-wave
                bf.y = ws[(kk + 1) * NT + nl];
                acc[nn] = __builtin_amdgcn_wmma_f32_16x16x4_f32(
                    false, a, false, bf, (short)0, acc[nn], false, false);
            }
        }
    }

    // ---- epilogue: D VGPR r holds M=r (lanes 0-15) / M=8+r (lanes 16-31); all N cols equal.
    if ((lane & 15) == 0) {
        const int bbase = b0 + ((lane >> 4) << 3);
#pragma unroll
        for (int nn = 0; nn < 4; ++nn) {
            const int n = n0 + (wave << 2) + nn;
#pragma unroll
            for (int r = 0; r < 8; ++r) {
                const size_t idx = (size_t)(bbase + r) * NNC + n;
                float vn = ALPHA * v[idx] + acc[nn][r] - VTH * z[idx];
                outv[idx] = vn;
                outz[idx] = (vn - VTH > 0.0f) ? 1.0f : 0.0f;
            }
        }
    }
}

extern "C" void kernel_launch(void* const* d_in, const int* in_sizes, int n_in,
                              void* d_out, int out_size, void* d_ws, size_t ws_size,
                              hipStream_t stream) {
    (void)in_sizes; (void)n_in; (void)d_ws; (void)ws_size; (void)out_size;
    const float* x = (const float*)d_in[0];   // (128, 1024, 512)
    const float* w = (const float*)d_in[1];   // (1024, 512)
    const float* v = (const float*)d_in[2];   // (128, 512)
    const float* z = (const float*)d_in[3];   // (128, 512)
    float* outv = (float*)d_out;              // v_new
    float* outz = outv + (size_t)BDIM * NNC;  // z_new (concatenated)

    dim3 grid(NNC / NT, BDIM / BT);           // 32 x 8 = 256 workgroups
    lif_wmma_kernel<<<grid, 128, 0, stream>>>(x, w, v, z, outv, outz);
}